// phi_6734508720267
// MI455X (gfx1250) — compile-verified
//
#include <hip/hip_runtime.h>
#include <hip/hip_bf16.h>

// ---------------------------------------------------------------------------
// Krylov expm-action on MI455X (gfx1250).
//   B=8, C=256, H=W=64, HW=4096, per-batch vec NB=2^20, total N=2^23 floats.
//   GEMM (fitness) uses V_WMMA_F32_16X16X4_F32 (f32-exact, matches reference).
//   All reductions are fixed-order two-stage trees (deterministic, no atomics).
// ---------------------------------------------------------------------------

#define BATCH   8
#define CDIM    256
#define HWDIM   4096
#define NB      (CDIM * HWDIM)          // 1<<20 per-batch elements
#define NB_LOG2 20
#define NTOT    ((size_t)BATCH * NB)    // 8,388,608
#define MKRY    8
#define NCHUNK  64                       // reduction chunks per batch

typedef float v2f __attribute__((ext_vector_type(2)));
typedef float v8f __attribute__((ext_vector_type(8)));

// ---------------------------------------------------------------- utilities
__global__ __launch_bounds__(256) void zero_kernel(float* p, int n) {
    int i = blockIdx.x * 256 + threadIdx.x;
    if (i < n) p[i] = 0.0f;
}

// partials[b*NCHUNK+chunk] = sum over chunk of x*y   (fixed-order tree)
__global__ __launch_bounds__(256) void dot_stage1(const float* __restrict__ x,
                                                  const float* __restrict__ y,
                                                  float* __restrict__ partials) {
    const int b = blockIdx.y, chunk = blockIdx.x;
    const size_t base = (size_t)b * NB + (size_t)chunk * (NB / NCHUNK);
    const float4* xp = (const float4*)(x + base);
    const float4* yp = (const float4*)(y + base);
    float s = 0.0f;
    for (int i = threadIdx.x; i < (NB / NCHUNK) / 4; i += 256) {
        float4 a = xp[i], c = yp[i];
        s += a.x * c.x + a.y * c.y + a.z * c.z + a.w * c.w;
    }
    __shared__ float red[256];
    red[threadIdx.x] = s;
    __syncthreads();
    for (int st = 128; st > 0; st >>= 1) {
        if (threadIdx.x < st) red[threadIdx.x] += red[threadIdx.x + st];
        __syncthreads();
    }
    if (threadIdx.x == 0) partials[b * NCHUNK + chunk] = red[0];
}

// out1[b*stride1] = (do_sqrt ? sqrt : id)(sum of partials[b,:])
__global__ __launch_bounds__(64) void dot_stage2(const float* __restrict__ partials,
                                                 float* __restrict__ out1, int stride1,
                                                 int do_sqrt) {
    const int b = blockIdx.x;
    __shared__ float red[64];
    red[threadIdx.x] = partials[b * NCHUNK + threadIdx.x];
    __syncthreads();
    for (int st = 32; st > 0; st >>= 1) {
        if (threadIdx.x < st) red[threadIdx.x] += red[threadIdx.x + st];
        __syncthreads();
    }
    if (threadIdx.x == 0) {
        float v = red[0];
        if (do_sqrt) v = sqrtf(v);
        out1[b * stride1] = v;
    }
}

// out = w / (h[b*hstride] + eps)
__global__ __launch_bounds__(256) void scale_kernel(const float* __restrict__ w,
                                                    const float* __restrict__ hptr, int hstride,
                                                    float eps, float* __restrict__ out) {
    size_t idx = (size_t)blockIdx.x * 256 + threadIdx.x;
    int b = (int)(idx >> NB_LOG2);
    out[idx] = w[idx] / (hptr[b * hstride] + eps);
}

// w -= h[b*hstride] * qj
__global__ __launch_bounds__(256) void axpy_kernel(float* __restrict__ w,
                                                   const float* __restrict__ qj,
                                                   const float* __restrict__ hptr, int hstride) {
    size_t idx = (size_t)blockIdx.x * 256 + threadIdx.x;
    int b = (int)(idx >> NB_LOG2);
    w[idx] -= hptr[b * hstride] * qj[idx];
}

// ------------------------------------------------------------- replicator
// r[b,c,hw] = s0*q - s0 * sum_c(s0*q).   One thread per (b,hw); coalesced
// across threads for each c; second pass hits L2 (per-block footprint ~512KB).
__global__ __launch_bounds__(256) void replicator_kernel(const float* __restrict__ s0,
                                                         const float* __restrict__ q,
                                                         float* __restrict__ r) {
    const int hw = blockIdx.x * 256 + threadIdx.x;
    const int b  = blockIdx.y;
    const float* sp = s0 + (size_t)b * NB + hw;
    const float* qp = q  + (size_t)b * NB + hw;
    float t = 0.0f;
#pragma unroll 4
    for (int c = 0; c < CDIM; ++c) t += sp[c * HWDIM] * qp[c * HWDIM];
    float* rp = r + (size_t)b * NB + hw;
#pragma unroll 4
    for (int c = 0; c < CDIM; ++c) {
        float s = sp[c * HWDIM];
        rp[c * HWDIM] = s * qp[c * HWDIM] - s * t;
    }
}

// ------------------------------------------------------------- WMMA GEMM
// Out[b,o,hw] = sum_c W[o,c] * R[b,c,hw].
// Grid (HW/512, C/16, B); block = 256 = 8 waves. Each wave: 16(M)x64(N) tile,
// K swept in steps of 4 via V_WMMA_F32_16X16X4_F32. W strip staged in LDS.
// Per ISA 7.12.2 (32-bit layouts, wave32):
//   A/B element v, lane l : K = v + 2*(l>=16)
//   C/D element v, lane l : M = v + 8*(l>=16), N = l%16
__global__ __launch_bounds__(256) void gemm_wmma(const float* __restrict__ Wm,
                                                 const float* __restrict__ R,
                                                 float* __restrict__ Out) {
    const int lane  = threadIdx.x & 31;
    const int wave  = threadIdx.x >> 5;
    const int b     = blockIdx.z;
    const int otile = blockIdx.y * 16;
    const int hwbase = (blockIdx.x * 8 + wave) * 64;

    __shared__ float Wlds[16 * CDIM];   // 16 KB
    for (int i = threadIdx.x; i < 16 * CDIM; i += 256) {
        int rr = i >> 8, cc = i & 255;
        Wlds[i] = Wm[(otile + rr) * CDIM + cc];
    }
    __syncthreads();

    const float* rb = R + (size_t)b * NB;
    const int mrow  = lane & 15;
    const int khalf = (lane >> 4) << 1;      // 0 or 2
    const int ncol  = hwbase + (lane & 15);

    v8f acc0 = {}, acc1 = {}, acc2 = {}, acc3 = {};

    for (int k = 0; k < CDIM; k += 4) {
        v2f a;
        a.x = Wlds[mrow * CDIM + k + khalf];
        a.y = Wlds[mrow * CDIM + k + khalf + 1];
        const float* p0 = rb + (size_t)(k + khalf) * HWDIM + ncol;
        const float* p1 = p0 + HWDIM;
        v2f b0; b0.x = p0[0];  b0.y = p1[0];
        v2f b1; b1.x = p0[16]; b1.y = p1[16];
        v2f b2; b2.x = p0[32]; b2.y = p1[32];
        v2f b3; b3.x = p0[48]; b3.y = p1[48];
        acc0 = __builtin_amdgcn_wmma_f32_16x16x4_f32(false, a, false, b0, (short)0, acc0, false, false);
        acc1 = __builtin_amdgcn_wmma_f32_16x16x4_f32(false, a, false, b1, (short)0, acc1, false, false);
        acc2 = __builtin_amdgcn_wmma_f32_16x16x4_f32(false, a, false, b2, (short)0, acc2, false, false);
        acc3 = __builtin_amdgcn_wmma_f32_16x16x4_f32(false, a, false, b3, (short)0, acc3, false, false);
    }

    float* ob = Out + (size_t)b * NB;
    const int rbase = otile + ((lane >> 4) << 3);
    const int cbase = hwbase + (lane & 15);
#pragma unroll
    for (int v = 0; v < 8; ++v) {
        ob[(size_t)(rbase + v) * HWDIM + cbase +  0] = acc0[v];
        ob[(size_t)(rbase + v) * HWDIM + cbase + 16] = acc1[v];
        ob[(size_t)(rbase + v) * HWDIM + cbase + 32] = acc2[v];
        ob[(size_t)(rbase + v) * HWDIM + cbase + 48] = acc3[v];
    }
}

// ------------------------------------------------------------- 9x9 expm
// ext = [[T_END*H(8x8), e1*T_END],[0,0]] (9x9); coords = expm(ext)[:8, 8].
// Scaling-and-squaring + 16-term Horner Taylor. One block per batch.
__global__ __launch_bounds__(128) void expm_kernel(const float* __restrict__ Hmat,
                                                   float* __restrict__ coords) {
    const int b = blockIdx.x;
    const int t = threadIdx.x;
    __shared__ float A[81], T1[81], T2[81];
    __shared__ int   sshift;
    if (t < 81) {
        int i = t / 9, j = t % 9;
        float v = 0.0f;
        if (i < 8 && j < 8) v = Hmat[b * 72 + i * 8 + j];   // T_END = 1
        if (i == 0 && j == 8) v = 1.0f;                     // T_END
        A[t] = v;
    }
    __syncthreads();
    if (t == 0) {
        float mx = 0.0f;
        for (int i = 0; i < 9; ++i) {
            float s = 0.0f;
            for (int j = 0; j < 9; ++j) s += fabsf(A[i * 9 + j]);
            mx = fmaxf(mx, s);
        }
        int sh = 0;
        while (mx > 0.5f && sh < 30) { mx *= 0.5f; ++sh; }
        sshift = sh;
    }
    __syncthreads();
    const int sh = sshift;
    if (t < 81) A[t] = ldexpf(A[t], -sh);
    __syncthreads();
    if (t < 81) T1[t] = (t / 9 == t % 9) ? 1.0f : 0.0f;
    __syncthreads();
    for (int kk = 16; kk >= 1; --kk) {            // Horner: M = I + (A*M)/kk
        if (t < 81) {
            int i = t / 9, j = t % 9;
            float acc = 0.0f;
            for (int c = 0; c < 9; ++c) acc += A[i * 9 + c] * T1[c * 9 + j];
            T2[t] = ((i == j) ? 1.0f : 0.0f) + acc / (float)kk;
        }
        __syncthreads();
        if (t < 81) T1[t] = T2[t];
        __syncthreads();
    }
    for (int q = 0; q < sh; ++q) {                // square back
        if (t < 81) {
            int i = t / 9, j = t % 9;
            float acc = 0.0f;
            for (int c = 0; c < 9; ++c) acc += T1[i * 9 + c] * T1[c * 9 + j];
            T2[t] = acc;
        }
        __syncthreads();
        if (t < 81) T1[t] = T2[t];
        __syncthreads();
    }
    if (t < 8) coords[b * 8 + t] = T1[t * 9 + 8];
}

// out = vnorm[b] * sum_j coords[b,j] * Q[j]
__global__ __launch_bounds__(256) void final_kernel(const float* __restrict__ Q,
                                                    const float* __restrict__ coords,
                                                    const float* __restrict__ vnorm,
                                                    float* __restrict__ out) {
    size_t idx = (size_t)blockIdx.x * 256 + threadIdx.x;
    int b = (int)(idx >> NB_LOG2);
    float acc = 0.0f;
#pragma unroll
    for (int j = 0; j < MKRY; ++j)
        acc += coords[b * 8 + j] * Q[(size_t)j * NTOT + idx];
    out[idx] = acc * vnorm[b];
}

// ---------------------------------------------------------------------------
extern "C" void kernel_launch(void* const* d_in, const int* in_sizes, int n_in,
                              void* d_out, int out_size, void* d_ws, size_t ws_size,
                              hipStream_t stream) {
    const float* s0 = (const float*)d_in[0];
    const float* v  = (const float*)d_in[1];
    const float* Wm = (const float*)d_in[2];
    float* out = (float*)d_out;

    float* ws       = (float*)d_ws;
    float* Q        = ws;                       // 8 * NTOT
    float* w        = Q + (size_t)MKRY * NTOT;  // NTOT
    float* r        = w + NTOT;                 // NTOT
    float* Hm       = r + NTOT;                 // B*9*8 = 576 (row-major [b,row,col])
    float* coords   = Hm + 576;                 // B*8
    float* vnorm    = coords + 64;              // B
    float* partials = vnorm + 8;                // B*NCHUNK

    const dim3 red1(NCHUNK, BATCH);
    const int  nblk = (int)(NTOT / 256);

    zero_kernel<<<3, 256, 0, stream>>>(Hm, 576);

    // vnorm = ||v||;  Q0 = v / vnorm
    dot_stage1<<<red1, 256, 0, stream>>>(v, v, partials);
    dot_stage2<<<BATCH, 64, 0, stream>>>(partials, vnorm, 1, /*sqrt=*/1);
    scale_kernel<<<nblk, 256, 0, stream>>>(v, vnorm, 1, 0.0f, Q);

    for (int k = 0; k < MKRY; ++k) {
        float* Qk = Q + (size_t)k * NTOT;
        // w = A(Qk) = W @ replicator(s0, Qk)
        replicator_kernel<<<dim3(HWDIM / 256, BATCH), 256, 0, stream>>>(s0, Qk, r);
        gemm_wmma<<<dim3(HWDIM / 512, CDIM / 16, BATCH), 256, 0, stream>>>(Wm, r, w);
        // modified Gram-Schmidt (sequential, matches reference)
        for (int j = 0; j <= k; ++j) {
            float* Qj = Q + (size_t)j * NTOT;
            dot_stage1<<<red1, 256, 0, stream>>>(Qj, w, partials);
            dot_stage2<<<BATCH, 64, 0, stream>>>(partials, Hm + j * 8 + k, 72, /*sqrt=*/0);
            axpy_kernel<<<nblk, 256, 0, stream>>>(w, Qj, Hm + j * 8 + k, 72);
        }
        if (k < MKRY - 1) {   // Q[8] / H[8,7] are dropped by the reference
            dot_stage1<<<red1, 256, 0, stream>>>(w, w, partials);
            dot_stage2<<<BATCH, 64, 0, stream>>>(partials, Hm + (k + 1) * 8 + k, 72, /*sqrt=*/1);
            scale_kernel<<<nblk, 256, 0, stream>>>(w, Hm + (k + 1) * 8 + k, 72, 1e-12f,
                                                   Q + (size_t)(k + 1) * NTOT);
        }
    }

    expm_kernel<<<BATCH, 128, 0, stream>>>(Hm, coords);
    final_kernel<<<nblk, 256, 0, stream>>>(Q, coords, vnorm, out);
}